// GraphAttentionBlock_2181843386766
// MI455X (gfx1250) — compile-verified
//
#include <hip/hip_runtime.h>
#include <hip/hip_bf16.h>

#define IN_DIM 128
#define HEADS 8
#define DHEAD 16
#define HD 128         // HEADS * DHEAD
#define LDS_PITCH 132  // 128 + 4 pad floats (TDM pad: conflict-free LDS banks)

typedef float v2f __attribute__((ext_vector_type(2)));
typedef float v8f __attribute__((ext_vector_type(8)));
typedef unsigned int u32x4 __attribute__((ext_vector_type(4)));
typedef int i32x4 __attribute__((ext_vector_type(4)));
typedef int i32x8 __attribute__((ext_vector_type(8)));

// ---------------------------------------------------------------------------
// Zero-fill kernel (d_out must start at 0 for the atomic scatter phase).
// ---------------------------------------------------------------------------
__global__ void zero_kernel(float* __restrict__ p, long n) {
    long i = (long)blockIdx.x * blockDim.x + threadIdx.x;
    long stride = (long)gridDim.x * blockDim.x;
    for (; i < n; i += stride) p[i] = 0.0f;
}

// ---------------------------------------------------------------------------
// Fused QKV projection GEMM using V_WMMA_F32_16X16X4_F32 (exact fp32 math).
// Block = 256 threads (8 wave32). Each block computes a 16-row x 128-col
// strip of one projection; wave w owns the 16x16 column tile at col0 = w*16.
// blockIdx.y selects the projection (0=Q, 1=K, 2=V).
//
// The 16x128 A strip (h rows) is shared by all 8 waves: it is staged into LDS
// once per block by the Tensor Data Mover, with TDM padding of 4 DWORDs every
// 128 DWORDs -> 132-float row pitch (conflict-free banks: 132 % 64 == 4).
// Because the TDM engine is the only writer of As (invisible to alias
// analysis), the A fragments are read with explicit ds_load_b64 inline asm;
// each asm block ends with s_wait_dscnt 0 so all consumers of its outputs
// execute strictly after the data has landed.
//
// A-frag layout (ISA 7.12.2, 32-bit A 16x4): lane L(<16) holds A[M=L,K=k0..k0+1]
// in v2f{x,y}; lane L+16 holds A[M=L,K=k0+2..k0+3].
// B-frag layout mirrors C/D row striping: lane n(<16) holds B[K=k0][N=n] /
// B[K=k0+1][N=n]; lane n+16 holds rows K=k0+2 / k0+3.
// C/D layout: lane L, VGPR r  ->  D[M = r + 8*(L/16), N = L%16].
// ---------------------------------------------------------------------------
__global__ void qkv_gemm_kernel(const float* __restrict__ h,
                                const float* __restrict__ Wq, const float* __restrict__ bq,
                                const float* __restrict__ Wk, const float* __restrict__ bk,
                                const float* __restrict__ Wv, const float* __restrict__ bv,
                                float* __restrict__ Qo, float* __restrict__ Ko,
                                float* __restrict__ Vo, int N)
{
    __shared__ float As[16][LDS_PITCH];

    // LDS byte offset of As: low 32 bits of the generic address (the shared
    // aperture tag lives in bits 63:32 per the ISA aperture check). Also
    // keeps the LDS allocation referenced/alive.
    const unsigned asBase = (unsigned)(size_t)&As[0][0];

    const float* W;
    const float* b;
    float* out;
    if (blockIdx.y == 0)      { W = Wq; b = bq; out = Qo; }
    else if (blockIdx.y == 1) { W = Wk; b = bk; out = Ko; }
    else                      { W = Wv; b = bv; out = Vo; }

    const int wave = threadIdx.x >> 5;   // 0..7 -> column tile
    const int lane = threadIdx.x & 31;
    const int lmod = lane & 15;
    const int lhi  = lane >> 4;          // 0 or 1

    const int row0 = blockIdx.x * 16;
    const int col0 = wave * 16;

    int rowsRem = N - row0;
    if (rowsRem > 16) rowsRem = 16;

    // ---- TDM: stage h[row0 .. row0+rowsRem) x 128 into LDS (wave 0 only) ----
    if (wave == 0) {
        const unsigned long long ga =
            (unsigned long long)(const void*)(h + (size_t)row0 * IN_DIM);

        u32x4 g0;
        g0.x = 1u;                                    // count=1 valid descriptor
        g0.y = asBase;                                // lds_addr (bytes)
        g0.z = (unsigned)(ga & 0xFFFFFFFFu);          // global_addr[31:0]
        g0.w = (unsigned)((ga >> 32) & 0x1FFFFFFu)    // global_addr[56:32]
             | (2u << 30);                            // type = 2 ("image")

        i32x8 g1;
        g1.s0 = (2 << 16)       // data_size = 2 -> 4-byte elements
              | (1 << 20)       // pad_enable
              | (6 << 22)       // pad_interval: 2^(6+1)=128 DWORDs
              | (3 << 25);      // pad_amount: 3+1 = 4 DWORDs
        g1.s1 = (IN_DIM & 0xFFFF) << 16;              // tensor_dim0 lo16 (=128)
        g1.s2 = (rowsRem & 0xFFFF) << 16;             // dim0 hi=0 | tensor_dim1 lo16
        g1.s3 = (IN_DIM << 16);                       // dim1 hi=0 | tile_dim0 = 128
        g1.s4 = rowsRem;                              // tile_dim1 | tile_dim2=0
        g1.s5 = IN_DIM;                               // tensor_dim0_stride lo32 (=128)
        g1.s6 = 0;                                    // stride0 hi | stride1 lo
        g1.s7 = 0;                                    // stride1 rest

        i32x4 gz4 = {0, 0, 0, 0};                     // groups 2/3 unused (2D)
        i32x8 gz8 = {0, 0, 0, 0, 0, 0, 0, 0};

        __builtin_amdgcn_tensor_load_to_lds(g0, g1, gz4, gz4, gz8, 0);
        __builtin_amdgcn_s_wait_tensorcnt(0);
    }
    __syncthreads();

    // ---- Preload all 32 A fragments (this lane's full K extent) from LDS ----
    // Address: As[lmod][2*lhi] ; fragment i covers K = 4*i (+16 bytes apart).
    v2f a[32];
    const unsigned lbase = asBase + (unsigned)((lmod * LDS_PITCH + 2 * lhi) * 4);

#define DS_LOAD8(o0,o1,o2,o3,o4,o5,o6,o7, i0,i1,i2,i3,i4,i5,i6,i7)            \
    asm volatile("ds_load_b64 %0, %8 offset:" #o0 "\n\t"                      \
                 "ds_load_b64 %1, %8 offset:" #o1 "\n\t"                      \
                 "ds_load_b64 %2, %8 offset:" #o2 "\n\t"                      \
                 "ds_load_b64 %3, %8 offset:" #o3 "\n\t"                      \
                 "ds_load_b64 %4, %8 offset:" #o4 "\n\t"                      \
                 "ds_load_b64 %5, %8 offset:" #o5 "\n\t"                      \
                 "ds_load_b64 %6, %8 offset:" #o6 "\n\t"                      \
                 "ds_load_b64 %7, %8 offset:" #o7 "\n\t"                      \
                 "s_wait_dscnt 0x0"                                           \
                 : "=v"(a[i0]), "=v"(a[i1]), "=v"(a[i2]), "=v"(a[i3]),        \
                   "=v"(a[i4]), "=v"(a[i5]), "=v"(a[i6]), "=v"(a[i7])         \
                 : "v"(lbase))

    DS_LOAD8(0,16,32,48,64,80,96,112,            0,1,2,3,4,5,6,7);
    DS_LOAD8(128,144,160,176,192,208,224,240,    8,9,10,11,12,13,14,15);
    DS_LOAD8(256,272,288,304,320,336,352,368,    16,17,18,19,20,21,22,23);
    DS_LOAD8(384,400,416,432,448,464,480,496,    24,25,26,27,28,29,30,31);
#undef DS_LOAD8

    const float* __restrict__ bcol = W + col0 + lmod;

    v8f acc = {0.f, 0.f, 0.f, 0.f, 0.f, 0.f, 0.f, 0.f};

    #pragma unroll
    for (int i = 0; i < 32; ++i) {
        const int k0 = 4 * i;
        v2f bb;
        bb.x = bcol[(size_t)(k0 + 2 * lhi) * HD];      // W[k0+2*lhi][col]
        bb.y = bcol[(size_t)(k0 + 2 * lhi + 1) * HD];  // W[k0+2*lhi+1][col]
        acc = __builtin_amdgcn_wmma_f32_16x16x4_f32(
            /*neg_a=*/false, a[i], /*neg_b=*/false, bb,
            /*c_mod=*/(short)0, acc, /*reuse_a=*/false, /*reuse_b=*/false);
    }

    const float bias = b[col0 + lmod];
    float* __restrict__ op = out + (size_t)row0 * HD + col0 + lmod;
    #pragma unroll
    for (int r = 0; r < 8; ++r) {
        const int M = r + 8 * lhi;
        if (M < rowsRem) op[(size_t)M * HD] = acc[r] + bias;
    }
}

// ---------------------------------------------------------------------------
// Edge phase: thread t -> edge e = t/8, head hh = t%8.
// score = dot16(K[src], Q[dst]) * 0.25 ; out[dst] += score * V[src]
// Q/K/V (76.8 MB total) are L2-resident on MI455X (192 MB L2), so the random
// gathers hit L2. Scatter via global_atomic_add_f32 (L2 atomic units).
// ---------------------------------------------------------------------------
__global__ void edge_attn_kernel(const float* __restrict__ Q,
                                 const float* __restrict__ K,
                                 const float* __restrict__ V,
                                 const int* __restrict__ src,
                                 const int* __restrict__ dst,
                                 float* __restrict__ out, int E)
{
    int t = blockIdx.x * blockDim.x + threadIdx.x;
    if (t >= E * HEADS) return;
    const int e  = t >> 3;
    const int hh = t & 7;
    const int s = src[e];
    const int d = dst[e];

    const float4* __restrict__ Kp = (const float4*)(K + (size_t)s * HD + hh * DHEAD);
    const float4* __restrict__ Qp = (const float4*)(Q + (size_t)d * HD + hh * DHEAD);
    const float4* __restrict__ Vp = (const float4*)(V + (size_t)s * HD + hh * DHEAD);

    // Prefetch the V row while the dot product is in flight (global_prefetch_b8).
    __builtin_prefetch((const void*)Vp, 0, 3);

    float sc = 0.0f;
    #pragma unroll
    for (int i = 0; i < 4; ++i) {
        const float4 kv = Kp[i];
        const float4 qv = Qp[i];
        sc += kv.x * qv.x + kv.y * qv.y + kv.z * qv.z + kv.w * qv.w;
    }
    sc *= 0.25f;  // 1/sqrt(D), D=16

    float* __restrict__ op = out + (size_t)d * HD + hh * DHEAD;
    #pragma unroll
    for (int i = 0; i < 4; ++i) {
        const float4 vv = Vp[i];
        atomicAdd(op + 4 * i + 0, vv.x * sc);
        atomicAdd(op + 4 * i + 1, vv.y * sc);
        atomicAdd(op + 4 * i + 2, vv.z * sc);
        atomicAdd(op + 4 * i + 3, vv.w * sc);
    }
}

// ---------------------------------------------------------------------------
// Host-side launch. Inputs (setup_inputs order):
//   0: h [N,128] f32   1: src [E] i32   2: dst [E] i32
//   3: Wq [128,128]    4: bq [128]      5: Wk   6: bk   7: Wv   8: bv
// Output: wV [N, 8, 16] f32 (flat N*128).
// ---------------------------------------------------------------------------
extern "C" void kernel_launch(void* const* d_in, const int* in_sizes, int n_in,
                              void* d_out, int out_size, void* d_ws, size_t ws_size,
                              hipStream_t stream) {
    const float* h   = (const float*)d_in[0];
    const int*   src = (const int*)d_in[1];
    const int*   dst = (const int*)d_in[2];
    const float* Wq  = (const float*)d_in[3];
    const float* bq  = (const float*)d_in[4];
    const float* Wk  = (const float*)d_in[5];
    const float* bk  = (const float*)d_in[6];
    const float* Wv  = (const float*)d_in[7];
    const float* bv  = (const float*)d_in[8];
    float* out = (float*)d_out;

    const int N = in_sizes[0] / IN_DIM;
    const int E = in_sizes[1];

    float* Qbuf = (float*)d_ws;
    float* Kbuf = Qbuf + (size_t)N * HD;
    float* Vbuf = Kbuf + (size_t)N * HD;

    // 1) Zero the output accumulator.
    {
        long n = (long)out_size;
        int blocks = (int)((n + 255 * 4) / (256 * 4));
        if (blocks > 4096) blocks = 4096;
        if (blocks < 1) blocks = 1;
        zero_kernel<<<blocks, 256, 0, stream>>>(out, n);
    }

    // 2) Fused Q/K/V projection GEMM (TDM-staged A + WMMA f32 16x16x4).
    {
        dim3 grid((N + 15) / 16, 3, 1);
        qkv_gemm_kernel<<<grid, 256, 0, stream>>>(h, Wq, bq, Wk, bk, Wv, bv,
                                                  Qbuf, Kbuf, Vbuf, N);
    }

    // 3) Edge-wise attention score + scatter-add.
    {
        long total = (long)E * HEADS;
        int blocks = (int)((total + 255) / 256);
        edge_attn_kernel<<<blocks, 256, 0, stream>>>(Qbuf, Kbuf, Vbuf, src, dst, out, E);
    }
}